// MultiHeadedAttention_798863917636
// MI455X (gfx1250) — compile-verified
//
#include <hip/hip_runtime.h>
#include <hip/hip_bf16.h>
#include <math.h>

// Problem dims (fixed by the reference)
#define D_DIM 512
#define T_DIM 512
#define V_DIM 512
#define F_DIM 12

typedef float v2f __attribute__((ext_vector_type(2)));
typedef float v8f __attribute__((ext_vector_type(8)));

// D = A(16x4, f32) x B(4x16, f32) + C(16x16, f32)
__device__ __forceinline__ v8f wmma_f32(v2f a, v2f b, v8f c) {
    return __builtin_amdgcn_wmma_f32_16x16x4_f32(
        /*neg_a=*/false, a, /*neg_b=*/false, b,
        /*c_mod=*/(short)0, c, /*reuse_a=*/false, /*reuse_b=*/false);
}

// C[M,N] = A[M,K] @ op(W) + bias, op(W) = W^T (TRANSB: W is [N,K]) or W ([K,N]).
// One 64x16 strip per wave (4 M-tiles share each B fragment), 8 waves/block.
// M multiple of 64, N multiple of 16, K multiple of 4.
// A-frag: lane(m + 16*(k>=2)), vgpr(k&1).  B-frag mirrored with n on lanes.
template <bool TRANSB>
__global__ __launch_bounds__(256) void gemm_wmma(
    const float* __restrict__ A, const float* __restrict__ W,
    const float* __restrict__ bias, float* __restrict__ C,
    int M, int N, int K)
{
    const int lane = threadIdx.x & 31;
    const int wave = threadIdx.x >> 5;
    const int lo   = lane & 15;       // m (for A) / n (for B/C)
    const int hi   = lane >> 4;       // 0: k=0,1   1: k=2,3
    const int tilesN = N >> 4;
    const int strip  = blockIdx.x * 8 + wave;
    const int si = strip / tilesN;        // 64-row strip index
    const int tj = strip - si * tilesN;   // 16-col tile index

    const int n = tj * 16 + lo;
    const float* a0    = A + (size_t)(si * 64 + lo) * K;
    const float* wrowT = W + (size_t)n * K;

    v8f c[4];
#pragma unroll
    for (int u = 0; u < 4; ++u) c[u] = (v8f){0.f,0.f,0.f,0.f,0.f,0.f,0.f,0.f};

    for (int k0 = 0; k0 < K; k0 += 4) {
        const int ka = k0 + 2 * hi;
        v2f b;
        if (TRANSB) {
            b.x = wrowT[ka];
            b.y = wrowT[ka + 1];
        } else {
            b.x = W[(size_t)ka * N + n];
            b.y = W[(size_t)(ka + 1) * N + n];
        }
#pragma unroll
        for (int u = 0; u < 4; ++u) {
            v2f a;
            a.x = a0[(size_t)(u * 16) * K + ka];
            a.y = a0[(size_t)(u * 16) * K + ka + 1];
            c[u] = wmma_f32(a, b, c[u]);
        }
    }
    const float bn = bias ? bias[n] : 0.0f;
#pragma unroll
    for (int u = 0; u < 4; ++u)
#pragma unroll
        for (int r = 0; r < 8; ++r)
            C[(size_t)(si * 64 + u * 16 + r + 8 * hi) * N + n] = c[u][r] + bn;
}

// bvo[i] = sum_k Wo[i,k] * bv[k]
__global__ void matvec_bvo(const float* __restrict__ Wo,
                           const float* __restrict__ bv,
                           float* __restrict__ bvo)
{
    int i = blockIdx.x * blockDim.x + threadIdx.x;
    if (i < D_DIM) {
        float s = 0.0f;
        for (int k = 0; k < D_DIM; ++k) s += Wo[(size_t)i * D_DIM + k] * bv[k];
        bvo[i] = s;
    }
}

#define ASTRIDE 516   // padded row stride (floats): column reads down 16 rows
                      // land on 16 distinct banks ((4*row + k) % 64 all distinct)

// One block per video v. Fused: logits -> softmax(F) -> weighted sum (+bo).
// LDS: ldsA = 16 rows x 516 (K_v / vo_v, rows 12..15 zeroed once) = 33.0 KB
//      ldsW = 12 rows x 512 (logits -> softmax weights)           = 24.6 KB
__global__ __launch_bounds__(256) void attn_kernel(
    const float* __restrict__ q,     // [T, D]
    const float* __restrict__ kbuf,  // [V, F, D]
    const float* __restrict__ vo,    // [V, F, D]  (already projected through Wo)
    const float* __restrict__ bo,    // [D]
    float* __restrict__ out)         // [V, T, D]
{
    __shared__ float ldsA[16 * ASTRIDE];
    __shared__ float ldsW[F_DIM * T_DIM];

    const int v    = blockIdx.x;
    const int tid  = threadIdx.x;
    const int lane = tid & 31;
    const int wave = tid >> 5;
    const int lo   = lane & 15;
    const int hi   = lane >> 4;

    // ---- Phase 1: stage K_v into LDS; zero padding rows 12..15 once ----
    const float* kv = kbuf + (size_t)v * F_DIM * D_DIM;
    for (int i = tid; i < F_DIM * D_DIM; i += 256) {
        const int f = i >> 9, d = i & 511;
        ldsA[f * ASTRIDE + d] = kv[i];
    }
    for (int i = tid; i < 4 * ASTRIDE; i += 256)
        ldsA[F_DIM * ASTRIDE + i] = 0.0f;
    __syncthreads();

    // ---- Phase 2: logits[f,t] = scale * dot(K_v[f,:], q[t,:]) via WMMA ----
    // k-loop outer; one A fragment (K_v) feeds 4 t-tile accumulators.
    const float scale = 0.044194173824159216f; // 1/sqrt(512)
    {
        const float* qrow[4];
#pragma unroll
        for (int j = 0; j < 4; ++j)
            qrow[j] = q + (size_t)((wave + 8 * j) * 16 + lo) * D_DIM;

        v8f c[4];
#pragma unroll
        for (int j = 0; j < 4; ++j) c[j] = (v8f){0.f,0.f,0.f,0.f,0.f,0.f,0.f,0.f};

        for (int k0 = 0; k0 < D_DIM; k0 += 4) {
            const int ka = k0 + 2 * hi;
            v2f a;
            a.x = ldsA[lo * ASTRIDE + ka];      // rows 12..15 are zeros
            a.y = ldsA[lo * ASTRIDE + ka + 1];
#pragma unroll
            for (int j = 0; j < 4; ++j) {
                v2f b;
                b.x = qrow[j][ka];
                b.y = qrow[j][ka + 1];
                c[j] = wmma_f32(a, b, c[j]);
            }
        }
#pragma unroll
        for (int j = 0; j < 4; ++j) {
            const int tt = wave + 8 * j;
#pragma unroll
            for (int r = 0; r < 8; ++r)
                if (r + 8 * hi < F_DIM)          // only the 12 real frame rows
                    ldsW[(r + 8 * hi) * T_DIM + tt * 16 + lo] = c[j][r] * scale;
        }
    }
    __syncthreads();

    // ---- Phase 3a: overwrite ldsA rows 0..11 with vo_v ----
    const float* vov = vo + (size_t)v * F_DIM * D_DIM;
    for (int i = tid; i < F_DIM * D_DIM; i += 256) {
        const int f = i >> 9, d = i & 511;
        ldsA[f * ASTRIDE + d] = vov[i];
    }
    // ---- Phase 3b: softmax over the 12 frames, per text column ----
    for (int t = tid; t < T_DIM; t += 256) {
        float l[F_DIM];
        float mx = -3.4e38f;
#pragma unroll
        for (int f = 0; f < F_DIM; ++f) {
            l[f] = ldsW[f * T_DIM + t];
            mx = fmaxf(mx, l[f]);
        }
        float s = 0.0f;
#pragma unroll
        for (int f = 0; f < F_DIM; ++f) { l[f] = expf(l[f] - mx); s += l[f]; }
        const float inv = 1.0f / s;
#pragma unroll
        for (int f = 0; f < F_DIM; ++f) ldsW[f * T_DIM + t] = l[f] * inv;
    }
    __syncthreads();

    // ---- Phase 4: out[v,t,d] = sum_f w[f,t] * vo_v[f,d] + bo[d] ----
    // A[m,k] = w[k, t0+m] (depends only on tt): hoist across 4 consecutive dd
    // tiles so each wave writes a contiguous 256B span of every output row.
    for (int tt = 0; tt < T_DIM / 16; ++tt) {
        v2f a[3];
#pragma unroll
        for (int s = 0; s < 3; ++s) {
            const int ka = 4 * s + 2 * hi;       // 0..11
            a[s].x = ldsW[ka * T_DIM + tt * 16 + lo];
            a[s].y = ldsW[(ka + 1) * T_DIM + tt * 16 + lo];
        }
#pragma unroll
        for (int j = 0; j < 4; ++j) {
            const int dd = wave * 4 + j;
            v8f c = {0.f,0.f,0.f,0.f,0.f,0.f,0.f,0.f};
#pragma unroll
            for (int s = 0; s < 3; ++s) {
                const int ka = 4 * s + 2 * hi;
                v2f b;
                b.x = ldsA[ka * ASTRIDE + dd * 16 + lo];
                b.y = ldsA[(ka + 1) * ASTRIDE + dd * 16 + lo];
                c = wmma_f32(a[s], b, c);
            }
            const float bias = bo[dd * 16 + lo];
            float* obase = out + ((size_t)v * T_DIM + tt * 16 + 8 * hi) * D_DIM
                               + dd * 16 + lo;
#pragma unroll
            for (int r = 0; r < 8; ++r)
                obase[(size_t)r * D_DIM] = c[r] + bias;
        }
    }
}

extern "C" void kernel_launch(void* const* d_in, const int* in_sizes, int n_in,
                              void* d_out, int out_size, void* d_ws, size_t ws_size,
                              hipStream_t stream) {
    const float* text  = (const float*)d_in[0];  // [T, D]
    const float* video = (const float*)d_in[1];  // [V, F, D]
    const float* Wq    = (const float*)d_in[2];
    const float* bq    = (const float*)d_in[3];
    const float* Wk    = (const float*)d_in[4];
    const float* bk    = (const float*)d_in[5];
    const float* Wv    = (const float*)d_in[6];
    const float* bv    = (const float*)d_in[7];
    const float* Wo    = (const float*)d_in[8];
    const float* bo    = (const float*)d_in[9];
    float* out = (float*)d_out;

    // Workspace carve-out (floats): q | k | Wvo | bvo | vo  (~27.3 MB total)
    float* ws   = (float*)d_ws;
    float* qb   = ws;                                  // 512*512
    float* kb   = qb  + (size_t)T_DIM * D_DIM;         // 6144*512
    float* Wvo  = kb  + (size_t)V_DIM * F_DIM * D_DIM; // 512*512
    float* bvo  = Wvo + (size_t)D_DIM * D_DIM;         // 512
    float* vob  = bvo + D_DIM;                         // 6144*512

    // q = text @ Wq^T + bq          (8x32 strips = 256 -> 32 blocks)
    gemm_wmma<true><<<32, 256, 0, stream>>>(text, Wq, bq, qb, T_DIM, D_DIM, D_DIM);
    // k = video @ Wk^T + bk         (96x32 strips = 3072 -> 384 blocks)
    gemm_wmma<true><<<384, 256, 0, stream>>>(video, Wk, bk, kb,
                                             V_DIM * F_DIM, D_DIM, D_DIM);
    // Wvo = Wo @ Wv  (so that vo = video @ Wvo^T + Wo@bv)
    gemm_wmma<false><<<32, 256, 0, stream>>>(Wo, Wv, nullptr, Wvo,
                                             D_DIM, D_DIM, D_DIM);
    matvec_bvo<<<2, 256, 0, stream>>>(Wo, bv, bvo);
    // vo = video @ Wvo^T + bvo
    gemm_wmma<true><<<384, 256, 0, stream>>>(video, Wvo, bvo, vob,
                                             V_DIM * F_DIM, D_DIM, D_DIM);
    // Fused logits -> softmax -> weighted sum (+bo), one block per video
    attn_kernel<<<V_DIM, 256, 0, stream>>>(qb, kb, vob, bo, out);
}